// CurrentEncode_57286273794981
// MI455X (gfx1250) — compile-verified
//
#include <hip/hip_runtime.h>

// LIF constant-current encoder, Norse defaults:
//   v_d = 0.9*v + 0.1*i
//   z   = heaviside(v_d - 1.0)
//   v'  = (1-z)*v_d            (V_RESET = 0)
//   i'  = 0.8*i + x
// Output: events[steps][c*h*w], float32.

#define LIF_STEPS 32
#define A_VMEM 0.9f   // 1 - DT*TAU_MEM_INV
#define B_VMEM 0.1f   // DT*TAU_MEM_INV
#define A_SYN  0.8f   // 1 - DT*TAU_SYN_INV
#define V_TH_C 1.0f

// Native Clang vector: works with __builtin_nontemporal_store and lowers to
// global_load_b128 / global_store_b128.
typedef float v4f __attribute__((ext_vector_type(4)));

__global__ __launch_bounds__(256) void CurrentEncode_57286273794981_kernel(
    const v4f* __restrict__ in, v4f* __restrict__ out, int n4)
{
    const int idx = blockIdx.x * blockDim.x + threadIdx.x;
    if (idx >= n4) return;

    // One coalesced 128-bit load of the constant input current.
    const v4f x = in[idx];

    v4f v = (v4f)(0.0f);
    v4f i = (v4f)(0.0f);

    v4f* outp = out + idx;

#pragma unroll
    for (int t = 0; t < LIF_STEPS; ++t) {
        // v_decayed = 0.9*v + 0.1*i   (uses OLD i)
        v4f vd;
        vd.x = fmaf(A_VMEM, v.x, B_VMEM * i.x);
        vd.y = fmaf(A_VMEM, v.y, B_VMEM * i.y);
        vd.z = fmaf(A_VMEM, v.z, B_VMEM * i.z);
        vd.w = fmaf(A_VMEM, v.w, B_VMEM * i.w);

        // spike + reset
        v4f z;
        z.x = (vd.x > V_TH_C) ? 1.0f : 0.0f;
        z.y = (vd.y > V_TH_C) ? 1.0f : 0.0f;
        z.z = (vd.z > V_TH_C) ? 1.0f : 0.0f;
        z.w = (vd.w > V_TH_C) ? 1.0f : 0.0f;

        v.x = (vd.x > V_TH_C) ? 0.0f : vd.x;
        v.y = (vd.y > V_TH_C) ? 0.0f : vd.y;
        v.z = (vd.z > V_TH_C) ? 0.0f : vd.z;
        v.w = (vd.w > V_TH_C) ? 0.0f : vd.w;

        // i' = 0.8*i + x
        i.x = fmaf(A_SYN, i.x, x.x);
        i.y = fmaf(A_SYN, i.y, x.y);
        i.z = fmaf(A_SYN, i.z, x.z);
        i.w = fmaf(A_SYN, i.w, x.w);

        // Streaming 128-bit store, nontemporal hint: output is write-once,
        // never re-read on-device -> keep it out of WGP$/L2 working set.
        __builtin_nontemporal_store(z, outp + (size_t)t * (size_t)n4);
    }
}

extern "C" void kernel_launch(void* const* d_in, const int* in_sizes, int n_in,
                              void* d_out, int out_size, void* d_ws, size_t ws_size,
                              hipStream_t stream) {
    (void)n_in; (void)out_size; (void)d_ws; (void)ws_size;

    const float* in_f = (const float*)d_in[0];
    float* out_f = (float*)d_out;

    const int n  = in_sizes[0];      // 3*640*640 = 1,228,800 (divisible by 4)
    const int n4 = n / 4;            // 307,200 v4f elements

    const int block = 256;           // 8 wave32s per block
    const int grid  = (n4 + block - 1) / block;

    CurrentEncode_57286273794981_kernel<<<grid, block, 0, stream>>>(
        (const v4f*)in_f, (v4f*)out_f, n4);
}